// AveragedHausdorffLoss_18872086299275
// MI455X (gfx1250) — compile-verified
//
#include <hip/hip_runtime.h>

typedef float v2f __attribute__((ext_vector_type(2)));
typedef float v8f __attribute__((ext_vector_type(8)));
typedef int   v4i __attribute__((ext_vector_type(4)));

#define N_PTS 8192
#define DIM   128
#define BM    128
#define BN    128
#define LDST  132   // 128 + 4 floats pad -> rows 4 banks apart, conflict-free frag reads

#if __has_builtin(__builtin_amdgcn_global_load_async_to_lds_b128)
#define USE_ASYNC_LDS 1
typedef __attribute__((address_space(1))) v4i g_v4i;   // global (AS1) int4
typedef __attribute__((address_space(3))) v4i l_v4i;   // LDS    (AS3) int4
#else
#define USE_ASYNC_LDS 0
#endif

// ---------------- kernel 0: init min buffers to +inf bit pattern ----------------
__global__ void hd_init_min(int* __restrict__ minBits) {
  int i = blockIdx.x * blockDim.x + threadIdx.x;
  if (i < 2 * N_PTS) minBits[i] = 0x7F800000;
}

// ---------------- kernel 1: squared norms, one wave32 per 128-dim row ----------------
__global__ __launch_bounds__(256)
void hd_norms(const float* __restrict__ X, const float* __restrict__ Y,
              float* __restrict__ x2, float* __restrict__ y2) {
  int wave = threadIdx.x >> 5;
  int lane = threadIdx.x & 31;
  int g = blockIdx.x * 8 + wave;                    // 0..16383
  const float* src = (g < N_PTS) ? X : Y;
  int r = (g < N_PTS) ? g : (g - N_PTS);
  const float4* row = (const float4*)(src + (size_t)r * DIM);
  float4 v = row[lane];                             // 32 lanes * 4 = 128 elems
  float s = v.x * v.x + v.y * v.y + v.z * v.z + v.w * v.w;
  #pragma unroll
  for (int m = 16; m >= 1; m >>= 1) s += __shfl_xor(s, m, 32);
  if (lane == 0) { if (g < N_PTS) x2[r] = s; else y2[r] = s; }
}

// ---------------- kernel 2: fused WMMA tile GEMM + distance + min epilogue ----------------
__global__ __launch_bounds__(256)
void hd_tile(const float* __restrict__ X, const float* __restrict__ Y,
             const float* __restrict__ x2, const float* __restrict__ y2,
             int* __restrict__ rowMinBits, int* __restrict__ colMinBits) {
  extern __shared__ float smem[];
  float* XS = smem;                                  // BM * LDST
  float* YS = XS + BM * LDST;                        // BN * LDST
  int*   colMinS = (int*)(YS + BN * LDST);           // BN ints

  const int tid  = threadIdx.x;
  const int wave = tid >> 5;
  const int lane = tid & 31;
  const int half = lane >> 4;
  const int l15  = lane & 15;
  const int bm   = blockIdx.x;
  const int bn   = blockIdx.y;

  if (tid < BN) colMinS[tid] = 0x7F800000;

  // cooperative tile load: 128 rows x 128 K fp32 each (64 KB per matrix)
  const float4* Xg = (const float4*)(X + (size_t)bm * BM * DIM);
  const float4* Yg = (const float4*)(Y + (size_t)bn * BN * DIM);
#if USE_ASYNC_LDS
  // CDNA5 async DMA path: memory -> LDS directly, tracked by ASYNCcnt,
  // no VGPR staging. 16B per lane per op, per-lane LDS addresses.
  #pragma unroll
  for (int k = 0; k < 16; ++k) {
    int l   = tid + k * 256;                         // 0..4095
    int row = l >> 5;
    int c4  = l & 31;
    __builtin_amdgcn_global_load_async_to_lds_b128(
        (g_v4i*)(Xg + row * 32 + c4), (l_v4i*)&XS[row * LDST + c4 * 4], 0, 0);
    __builtin_amdgcn_global_load_async_to_lds_b128(
        (g_v4i*)(Yg + row * 32 + c4), (l_v4i*)&YS[row * LDST + c4 * 4], 0, 0);
  }
#if __has_builtin(__builtin_amdgcn_s_wait_asynccnt)
  __builtin_amdgcn_s_wait_asynccnt(0);
#else
  asm volatile("s_wait_asynccnt 0" ::: "memory");
#endif
#else
  #pragma unroll
  for (int k = 0; k < 16; ++k) {
    int l   = tid + k * 256;                         // 0..4095
    int row = l >> 5;
    int c4  = l & 31;
    float4 vx = Xg[row * 32 + c4];
    float4 vy = Yg[row * 32 + c4];
    *(float4*)&XS[row * LDST + c4 * 4] = vx;         // 528B row stride, 16B aligned
    *(float4*)&YS[row * LDST + c4 * 4] = vy;
  }
#endif
  __syncthreads();

  // wave w computes rows [16w,16w+16) x all 128 cols : 8 x v8f accumulators
  v8f acc[8];
  #pragma unroll
  for (int n = 0; n < 8; ++n) acc[n] = (v8f){};

  const int arow = (wave * 16 + l15) * LDST;
  #pragma unroll 4
  for (int k = 0; k < DIM; k += 4) {
    // fp32 16x4 A fragment: VGPR0 = K{k | k+2}, VGPR1 = K{k+1 | k+3} across lane halves
    v2f a = *(const v2f*)&XS[arow + k + 2 * half];
    #pragma unroll
    for (int n = 0; n < 8; ++n) {
      v2f b = *(const v2f*)&YS[(n * 16 + l15) * LDST + k + 2 * half];
      acc[n] = __builtin_amdgcn_wmma_f32_16x16x4_f32(
          false, a, false, b, (short)0, acc[n], false, false);
    }
  }

  // epilogue: distances + row/col minima (dist >= 0 => int compare == float compare)
  float x2r[8], y2c[8];
  #pragma unroll
  for (int e = 0; e < 8; ++e) x2r[e] = x2[bm * BM + wave * 16 + half * 8 + e];
  #pragma unroll
  for (int n = 0; n < 8; ++n) y2c[n] = y2[bn * BN + n * 16 + l15];

  float rmin[8];
  #pragma unroll
  for (int e = 0; e < 8; ++e) rmin[e] = __builtin_inff();

  #pragma unroll
  for (int n = 0; n < 8; ++n) {
    float cmin = __builtin_inff();
    #pragma unroll
    for (int e = 0; e < 8; ++e) {
      float sq = x2r[e] + y2c[n] - 2.0f * acc[n][e];
      float d  = sqrtf(fmaxf(sq, 0.0f));
      rmin[e]  = fminf(rmin[e], d);
      cmin     = fminf(cmin, d);
    }
    cmin = fminf(cmin, __shfl_xor(cmin, 16, 32));    // merge the two lane halves
    if (half == 0) atomicMin(&colMinS[n * 16 + l15], __float_as_int(cmin));
  }

  #pragma unroll
  for (int e = 0; e < 8; ++e) {
    float r = rmin[e];
    r = fminf(r, __shfl_xor(r, 1, 32));
    r = fminf(r, __shfl_xor(r, 2, 32));
    r = fminf(r, __shfl_xor(r, 4, 32));
    r = fminf(r, __shfl_xor(r, 8, 32));
    if (l15 == 0)
      atomicMin(&rowMinBits[bm * BM + wave * 16 + half * 8 + e], __float_as_int(r));
  }

  __syncthreads();
  if (tid < BN) atomicMin(&colMinBits[bn * BN + tid], colMinS[tid]);
}

// ---------------- kernel 3: final sum of 16384 mins, *0.5 ----------------
__global__ __launch_bounds__(256)
void hd_reduce(const int* __restrict__ minBits, float* __restrict__ out) {
  __shared__ float red[256];
  float s = 0.0f;
  for (int i = threadIdx.x; i < 2 * N_PTS; i += 256) s += __int_as_float(minBits[i]);
  red[threadIdx.x] = s;
  __syncthreads();
  for (int off = 128; off > 0; off >>= 1) {
    if (threadIdx.x < off) red[threadIdx.x] += red[threadIdx.x + off];
    __syncthreads();
  }
  if (threadIdx.x == 0) out[0] = red[0] * 0.5f;
}

extern "C" void kernel_launch(void* const* d_in, const int* in_sizes, int n_in,
                              void* d_out, int out_size, void* d_ws, size_t ws_size,
                              hipStream_t stream) {
  (void)in_sizes; (void)n_in; (void)out_size; (void)ws_size;
  const float* X = (const float*)d_in[0];
  const float* Y = (const float*)d_in[1];

  float* x2 = (float*)d_ws;                 // 8192 f32
  float* y2 = x2 + N_PTS;                   // 8192 f32
  int* minBits = (int*)(y2 + N_PTS);        // 16384 i32: [rowMin | colMin]
  int* rowMinBits = minBits;
  int* colMinBits = minBits + N_PTS;

  hd_init_min<<<(2 * N_PTS + 255) / 256, 256, 0, stream>>>(minBits);
  hd_norms<<<(2 * N_PTS) / 8, 256, 0, stream>>>(X, Y, x2, y2);

  dim3 grid(N_PTS / BM, N_PTS / BN);        // 64 x 64 tiles
  size_t smem = (size_t)(2 * BM * LDST) * sizeof(float) + BN * sizeof(int);
  hd_tile<<<grid, 256, smem, stream>>>(X, Y, x2, y2, rowMinBits, colMinBits);

  hd_reduce<<<1, 256, 0, stream>>>(minBits, (float*)d_out);
}